// IntLogLik_6116033429509
// MI455X (gfx1250) — compile-verified
//
#include <hip/hip_runtime.h>
#include <hip/hip_bf16.h>
#include <stdint.h>

// ---------------------------------------------------------------------------
// IntLogLik for MI455X (gfx1250).
// Per location: forward-substitution solve of a 64x64 lower-triangular system,
// ||x||^2, sum(log(diag)), + uniform scalar tail.
// Bandwidth-bound (~0.5 FLOP/byte): optimize the data path, not FLOPs.
//  - GChol staged global->LDS with CDNA5 async copies (ASYNCcnt), double
//    buffered per wave; only the 16B chunks covering the lower triangle are
//    loaded (544 chunks = 17 wave32 B128 instructions per location, ~47%
//    traffic cut vs dense).
//  - wave32: one wave per location, lane owns rows {l, l+32}; L rows are
//    copied LDS->VGPRs (ds_load_b128) so the 64-step solve chain is pure
//    v_readlane + v_fma.
// ---------------------------------------------------------------------------

#define NUG_MULT_F       4.0f
#define NREP             64
#define LSTRIDE          68                       // floats per LDS row (272B: 16B aligned, <=2-way bank conflicts)
#define BUF_FLOATS       (NREP * LSTRIDE)         // 4352 floats = 17408 B per buffer
#define WAVES_PER_BLOCK  4
#define THREADS          (WAVES_PER_BLOCK * 32)
#define NCHUNK           17                       // 544 16B lower-tri chunks / 32 lanes
#define SMEM_BYTES       (WAVES_PER_BLOCK * 2 * BUF_FLOATS * 4)   // 139,264 B (<320KB WGP LDS)
#define MAX_BLOCKS       2048

// uniform-index broadcast via v_readlane (srclane is wave-uniform here)
__device__ __forceinline__ float bcast(float v, int srclane) {
  return __uint_as_float(__builtin_amdgcn_readlane(__float_as_uint(v), srclane));
}

// CDNA5 async global -> LDS copy, 16B per lane. Tracked by ASYNCcnt.
__device__ __forceinline__ void async_ld_b128(uint32_t lds_byte_addr, const void* gaddr) {
  asm volatile("global_load_async_to_lds_b128 %0, %1, off"
               :: "v"(lds_byte_addr), "v"(gaddr)
               : "memory");
}

__device__ __forceinline__ void wait_async0() {
  asm volatile("s_wait_asynccnt 0" ::: "memory");
}

__global__ __launch_bounds__(THREADS)
void intloglik_kernel(const float* __restrict__ resp,     // (64, N)
                      const float* __restrict__ gchol,    // (N, 64, 64) lower-tri
                      const float* __restrict__ nug_mean, // (N,)
                      float* __restrict__ out,            // (N,)
                      int N)
{
  extern __shared__ float smem[];
  const int lane = threadIdx.x & 31;
  const int wib  = threadIdx.x >> 5;                       // wave in block
  const int wave = blockIdx.x * WAVES_PER_BLOCK + wib;     // global wave id
  const int nw   = gridDim.x * WAVES_PER_BLOCK;            // total waves

  const int boff = wib * 2 * BUF_FLOATS;                   // this wave's buffer pair (floats)
  // LDS byte address of smem start (generic LDS ptr: low 32 bits = LDS offset)
  const uint32_t smem_lds = (uint32_t)(uintptr_t)(void*)smem;
  uint32_t lbase[2];
  lbase[0] = smem_lds + (uint32_t)(boff * 4);
  lbase[1] = lbase[0] + (uint32_t)(BUF_FLOATS * 4);

  // ------------------------------------------------------------------
  // Per-lane (global, LDS) byte offsets covering the lower triangle in
  // 16B chunks. Row r needs chunks 0..floor(r/4); group g = rows
  // 4g..4g+3, each with g+1 chunks; cum chunks before group g = 2g(g+1).
  // Total = 2*16*17 = 544 = 17 * 32 lanes exactly. Branchless selects.
  // ------------------------------------------------------------------
  uint32_t goffs[NCHUNK], loffs[NCHUNK];
#pragma unroll
  for (int i = 0; i < NCHUNK; ++i) {
    const int c = i * 32 + lane;                            // 0..543
    int g = 0;
#pragma unroll
    for (int t = 1; t <= 15; ++t) g = (c >= 2 * t * (t + 1)) ? t : g;
    const int rem = c - 2 * g * (g + 1);
    const int gp1 = g + 1;
    const int rig = (rem >= 3 * gp1) ? 3 : (rem >= 2 * gp1) ? 2 : (rem >= gp1) ? 1 : 0;
    const int q   = rem - rig * gp1;                        // chunk within row
    const int r   = 4 * g + rig;                            // row
    goffs[i] = (uint32_t)(r * (NREP * 4)    + q * 16);
    loffs[i] = (uint32_t)(r * (LSTRIDE * 4) + q * 16);
  }

  // uniform constants (alpha does not depend on nug_mean for this model)
  const float alpha      = 1.0f / (NUG_MULT_F * NUG_MULT_F) + 2.0f;  // 2.0625
  const float alpha_post = alpha + 0.5f * (float)NREP;               // 34.0625
  const float lg_const   = lgammaf(alpha_post) - lgammaf(alpha);     // hoisted, cold

  // ------------------------------------------------------------------
  // Persistent loop over locations, double-buffered async pipeline:
  //   wait(cur) -> issue(next) -> compute(cur)
  // ------------------------------------------------------------------
  int loc = wave;
  if (loc < N) {
    const char* gl = (const char*)gchol + (size_t)loc * (NREP * NREP * 4);
#pragma unroll
    for (int i = 0; i < NCHUNK; ++i) async_ld_b128(lbase[0] + loffs[i], gl + goffs[i]);
  }

  int p = 0;
  for (; loc < N; loc += nw, p ^= 1) {
    // RHS + nug loads first: global latency overlaps the async LDS fill
    float b0 = resp[(size_t)lane * N + loc];
    float b1 = resp[(size_t)(lane + 32) * N + loc];
    const float nug = nug_mean[loc];

    wait_async0();                                           // buffer p ready

    const int nxt = loc + nw;
    if (nxt < N) {                                           // prefetch next location
      const char* gl = (const char*)gchol + (size_t)nxt * (NREP * NREP * 4);
      const uint32_t lb = lbase[p ^ 1];
#pragma unroll
      for (int i = 0; i < NCHUNK; ++i) async_ld_b128(lb + loffs[i], gl + goffs[i]);
    }

    // integer buffer base -> direct smem[] indexing keeps AS(3) provable
    const int B = boff + (p ? BUF_FLOATS : 0);

    // ---- copy this lane's L rows LDS -> registers (ds_load_b128 x24) ----
    float A0[32], A1[32], A2[32];   // row lane cols 0..31; row lane+32 cols 0..31; cols 32..63
#pragma unroll
    for (int k = 0; k < 8; ++k) {
      const float4 t0 = *(const float4*)&smem[B + lane * LSTRIDE + 4 * k];
      const float4 t1 = *(const float4*)&smem[B + (lane + 32) * LSTRIDE + 4 * k];
      const float4 t2 = *(const float4*)&smem[B + (lane + 32) * LSTRIDE + 32 + 4 * k];
      A0[4*k+0] = t0.x; A0[4*k+1] = t0.y; A0[4*k+2] = t0.z; A0[4*k+3] = t0.w;
      A1[4*k+0] = t1.x; A1[4*k+1] = t1.y; A1[4*k+2] = t1.z; A1[4*k+3] = t1.w;
      A2[4*k+0] = t2.x; A2[4*k+1] = t2.y; A2[4*k+2] = t2.z; A2[4*k+3] = t2.w;
    }

    // diagonal reciprocals + log-det contribution (off the critical chain)
    const float d0 = smem[B + lane * LSTRIDE + lane];
    const float d1 = smem[B + (lane + 32) * LSTRIDE + (lane + 32)];
    const float inv0 = 1.0f / d0;
    const float inv1 = 1.0f / d1;
    float ldet = logf(d0) + logf(d1);

    float sum2 = 0.0f;

    // forward substitution, columns 0..31 (pivot rows live in b0)
#pragma unroll
    for (int j = 0; j < 32; ++j) {
      const float xj = bcast(b0, j) * bcast(inv0, j);
      sum2 = fmaf(xj, xj, sum2);
      if (lane > j) b0 = fmaf(-A0[j], xj, b0);               // above-diag lanes masked
      b1 = fmaf(-A1[j], xj, b1);                             // row lane+32 > j always
    }
    // columns 32..63 (pivot rows live in b1; b0 rows are all done)
#pragma unroll
    for (int j = 0; j < 32; ++j) {
      const float xj = bcast(b1, j) * bcast(inv1, j);
      sum2 = fmaf(xj, xj, sum2);
      if (lane > j) b1 = fmaf(-A2[j], xj, b1);
    }

    // reduce logdet across the wave (sum2 is already wave-uniform)
#pragma unroll
    for (int off = 16; off > 0; off >>= 1)
      ldet += __shfl_xor(ldet, off, 32);

    if (lane == 0) {
      const float beta      = nug * (alpha - 1.0f);
      const float beta_post = beta + 0.5f * sum2;
      out[loc] = -ldet + alpha * logf(beta) - alpha_post * logf(beta_post) + lg_const;
    }
  }
}

extern "C" void kernel_launch(void* const* d_in, const int* in_sizes, int n_in,
                              void* d_out, int out_size, void* d_ws, size_t ws_size,
                              hipStream_t stream) {
  const float* resp  = (const float*)d_in[0];   // (64, N)
  const float* gchol = (const float*)d_in[1];   // (N, 64, 64)
  const float* nug   = (const float*)d_in[2];   // (N,)
  float* out = (float*)d_out;
  const int N = in_sizes[2];

  int nb = (N + WAVES_PER_BLOCK - 1) / WAVES_PER_BLOCK;
  if (nb > MAX_BLOCKS) nb = MAX_BLOCKS;         // persistent waves -> async pipeline depth > 1
  if (nb < 1) nb = 1;

  intloglik_kernel<<<nb, THREADS, SMEM_BYTES, stream>>>(resp, gchol, nug, out, N);
}